// GaussianRegLoss_47785806135451
// MI455X (gfx1250) — compile-verified
//
#include <hip/hip_runtime.h>
#include <math.h>

typedef float v2f __attribute__((ext_vector_type(2)));
typedef float v8f __attribute__((ext_vector_type(8)));

#define NJ 23
#define NPOS (64*64*64)                              // 262144 positions per (b,j)
#define BS 4
#define THREADS 256
#define WAVES_PER_BLOCK (THREADS/32)                 // 8
#define BLOCKS_PER_BATCH 256
#define TILES_PER_BATCH (NPOS/16)                    // 16384 position-tiles of 16
#define TILES_PER_BLOCK (TILES_PER_BATCH/BLOCKS_PER_BATCH)   // 64
#define TILES_PER_WAVE (TILES_PER_BLOCK/WAVES_PER_BLOCK)     // 8
#define TOTAL_BLOCKS (BS*BLOCKS_PER_BATCH)           // 1024

#define LOG2E 1.4426950408889634f

#if __has_builtin(__builtin_amdgcn_exp2f)
#define EXP2F(x) __builtin_amdgcn_exp2f(x)
#else
#define EXP2F(x) exp2f(x)
#endif

__device__ __forceinline__ float fast_sigmoid(float x) {
#if __has_builtin(__builtin_amdgcn_tanhf)
    return fmaf(0.5f, __builtin_amdgcn_tanhf(x * 0.5f), 0.5f);
#else
    return __builtin_amdgcn_rcpf(1.f + EXP2F(x * -LOG2E));
#endif
}

// One block = one batch slab; each wave processes 16-position tiles.
// Per tile: 2x V_WMMA_F32_16X16X4_F32 give D = log2e*(2*g.c - |g|^2 - |c|^2)
// = -sq_dist*log2e, so exp2(D) = exp(-sq_dist) directly.
__global__ void gauss_loss_main(const float* __restrict__ kpts_pred,
                                const float* __restrict__ heatmaps,
                                const float* __restrict__ grids,
                                float* __restrict__ partials)
{
    const int b    = blockIdx.x / BLOCKS_PER_BATCH;
    const int blk  = blockIdx.x % BLOCKS_PER_BATCH;
    const int tid  = threadIdx.x;
    const int wave = tid >> 5;
    const int lane = tid & 31;
    const int ln   = lane & 15;        // position index within tile / A row
    const int hi   = lane >> 4;        // lane half: K pair (A/B), M half (C/D)

    const float* kp = kpts_pred + b * 3 * NJ;
    const float* hb = heatmaps + (size_t)b * NJ * NPOS;
    const float* gb = grids    + (size_t)b * NPOS * 3;

    // Loop-invariant lane state.
    //   A[t]      : 2*log2e*kpts for joint tile t (A layout, K=3 col = 0)
    //   ncn[t][r] : -log2e*|c_j|^2 for the joint behind D VGPR r in this lane
    //   msk1[r]   : validity of tile-1 joints (tile 0 always valid)
    //   off*[r]   : clamped 32-bit heatmap offsets (j*NPOS + ln)
    v2f      A[2];
    float    ncn[2][8];
    float    msk1[7];
    unsigned off0[8];
    unsigned off1[7];
    #pragma unroll
    for (int t = 0; t < 2; ++t) {
        const int j = t*16 + ln;
        const bool v = (j < NJ);
        const float kx = v ? kp[0*NJ + j] : 0.f;
        const float ky = v ? kp[1*NJ + j] : 0.f;
        const float kz = v ? kp[2*NJ + j] : 0.f;
        A[t].x = (2.f * LOG2E) * (hi ? kz : kx);   // K=0 (x) / K=2 (z)
        A[t].y = (2.f * LOG2E) * (hi ? 0.f : ky);  // K=1 (y) / K=3 (pad)
        #pragma unroll
        for (int r = 0; r < 8; ++r) {
            const int  jr = t*16 + r + 8*hi;
            const bool vr = (jr < NJ);
            const int  jc = vr ? jr : (NJ - 1);    // clamp -> valid address
            const float cx = kp[0*NJ + jc];
            const float cy = kp[1*NJ + jc];
            const float cz = kp[2*NJ + jc];
            ncn[t][r] = vr ? (-LOG2E * (cx*cx + cy*cy + cz*cz)) : 0.f;
            if (t == 0) {
                off0[r] = (unsigned)(jc * NPOS + ln);
            } else if (r < 7) {
                msk1[r] = vr ? 1.f : 0.f;
                off1[r] = (unsigned)(jc * NPOS + ln);
            }
            // (t==1, r==7) is invalid for BOTH lane halves -> dropped entirely
        }
    }

    float sum = 0.f;
    const int tile0 = blk * TILES_PER_BLOCK + wave * TILES_PER_WAVE;
    for (int i = 0; i < TILES_PER_WAVE; ++i) {
        const unsigned P0 = (unsigned)(tile0 + i) * 16u;
        const unsigned p  = P0 + (unsigned)ln;

        // ---- issue ALL loads up front (unconditional, batched) ----
        const float gx = __builtin_nontemporal_load(gb + p*3u + 0u);
        const float gy = __builtin_nontemporal_load(gb + p*3u + 1u);
        const float gz = __builtin_nontemporal_load(gb + p*3u + 2u);
        float hm0[8], hm1[7];
        #pragma unroll
        for (int r = 0; r < 8; ++r)
            hm0[r] = __builtin_nontemporal_load(hb + (off0[r] + P0));
        #pragma unroll
        for (int r = 0; r < 7; ++r)
            hm1[r] = __builtin_nontemporal_load(hb + (off1[r] + P0));

        // ---- matrix part: D = -sq_dist * log2e ----
        const float gn2 = LOG2E * (gx*gx + gy*gy + gz*gz);
        v2f B;
        B.x = hi ? gz : gx;            // K=0 / K=2
        B.y = hi ? 0.f : gy;           // K=1 / K=3 (pad)
        v8f d[2];
        #pragma unroll
        for (int t = 0; t < 2; ++t) {
            v8f c;
            #pragma unroll
            for (int r = 0; r < 8; ++r) c[r] = ncn[t][r] - gn2;  // plain sub, no xor
            d[t] = __builtin_amdgcn_wmma_f32_16x16x4_f32(
                       false, A[t], false, B, (short)0, c, false, false);
        }

        // ---- elementwise loss: e = 0.02*exp2(D) - sigmoid(hm) ----
        #pragma unroll
        for (int r = 0; r < 8; ++r) {              // tile 0: all joints valid
            const float s     = fast_sigmoid(hm0[r]);
            const float gauss = EXP2F(d[0][r]);
            const float e     = fmaf(0.02f, gauss, -s);
            sum = fmaf(e, e, sum);
        }
        #pragma unroll
        for (int r = 0; r < 7; ++r) {              // tile 1: mask pad joints
            const float s     = fast_sigmoid(hm1[r]);
            const float gauss = EXP2F(d[1][r]);
            const float e     = fmaf(0.02f, gauss, -s) * msk1[r];
            sum = fmaf(e, e, sum);
        }
    }

    // ---- wave reduction (shuffle tree), then cross-wave via 8 LDS slots ----
    #pragma unroll
    for (int k = 16; k > 0; k >>= 1)
        sum += __shfl_xor(sum, k, 32);

    __shared__ float wsum[WAVES_PER_BLOCK];
    if (lane == 0) wsum[wave] = sum;
    __syncthreads();
    if (tid == 0) {
        float t = 0.f;
        #pragma unroll
        for (int w = 0; w < WAVES_PER_BLOCK; ++w) t += wsum[w];
        partials[blockIdx.x] = t;
    }
}

// Final deterministic reduction of per-block partials -> mean
__global__ void gauss_loss_reduce(const float* __restrict__ partials,
                                  float* __restrict__ out)
{
    __shared__ float red[256];
    const int tid = threadIdx.x;
    float s = 0.f;
    for (int i = tid; i < TOTAL_BLOCKS; i += 256) s += partials[i];
    red[tid] = s;
    __syncthreads();
    #pragma unroll
    for (int k = 128; k > 0; k >>= 1) {
        if (tid < k) red[tid] += red[tid + k];
        __syncthreads();
    }
    if (tid == 0)
        out[0] = red[0] * (1.0f / ((float)BS * (float)NJ * (float)NPOS));
}

extern "C" void kernel_launch(void* const* d_in, const int* in_sizes, int n_in,
                              void* d_out, int out_size, void* d_ws, size_t ws_size,
                              hipStream_t stream)
{
    // setup_inputs order: kpts_gt(0, unused), kpts_pred(1), heatmaps(2), grids(3)
    const float* kpts_pred = (const float*)d_in[1];
    const float* heatmaps  = (const float*)d_in[2];
    const float* grids     = (const float*)d_in[3];
    float* partials = (float*)d_ws;   // TOTAL_BLOCKS floats = 4 KB

    gauss_loss_main<<<dim3(TOTAL_BLOCKS), dim3(THREADS), 0, stream>>>(
        kpts_pred, heatmaps, grids, partials);
    gauss_loss_reduce<<<dim3(1), dim3(256), 0, stream>>>(partials, (float*)d_out);
}